// PACE_12403865550926
// MI455X (gfx1250) — compile-verified
//
#include <hip/hip_runtime.h>
#include <hip/hip_bf16.h>
#include <math.h>

typedef float v2f __attribute__((ext_vector_type(2)));
typedef float v8f __attribute__((ext_vector_type(8)));

#define NPT   100            // N points
#define KK    10             // K shapes
#define P3N   300            // 3*N
#define MR    310            // rows of M (+h)
#define MRP   320            // padded rows (20 tiles of 16)
#define KPAD  312            // padded contraction dim (78 steps of 4)

// workspace layout (floats)
#define M_OFF    0                       // Mpad [320][312]
#define H_OFF    (M_OFF + MRP*KPAD)      // h    [320]
#define GBT_OFF  (H_OFF + MRP)           // GBt  [10][300]
#define G_OFF    (GBT_OFF + KK*P3N)      // g    [10]
#define BW_OFF   (G_OFF + KK)            // b_w  [10][3]
#define SW_OFF   (BW_OFF + 3*KK)         // sw   [100]
#define WSUM_OFF (SW_OFF + NPT)          // wsum

#define WAVES 2

// ---------------------------------------------------------------------------
// Kernel 1: batch-independent precompute (single workgroup)
// ---------------------------------------------------------------------------
__global__ void pace_precompute(const float* __restrict__ w,
                                const float* __restrict__ bkp,
                                float* __restrict__ ws) {
  __shared__ float sBarB[P3N * KK];   // barB[p][k]
  __shared__ float sGBt[KK * P3N];    // GBt[k][p]
  __shared__ float sG[KK * KK];
  __shared__ float sHinv[KK * KK];
  __shared__ float sH[KK * KK];
  __shared__ float sbw[3 * KK];       // b_w[k][i] at k*3+i
  __shared__ float ssw[NPT];
  __shared__ float sg[KK];
  __shared__ float swsum;

  const int t = threadIdx.x, nt = blockDim.x;

  for (int n = t; n < NPT; n += nt) ssw[n] = sqrtf(w[n]);
  if (t == 0) { float s = 0.f; for (int n = 0; n < NPT; ++n) s += w[n]; swsum = s; }
  __syncthreads();

  for (int e = t; e < 3 * KK; e += nt) {
    int k = e / 3, i = e % 3;
    float s = 0.f;
    for (int n = 0; n < NPT; ++n) s += bkp[k * P3N + i * NPT + n] * w[n];
    sbw[e] = s / swsum;
  }
  __syncthreads();

  // barB[3n+i][k] = sw[n]*(b[k,i,n]-b_w[k,i])
  for (int e = t; e < P3N * KK; e += nt) {
    int p = e / KK, k = e % KK, n = p / 3, i = p % 3;
    sBarB[e] = ssw[n] * (bkp[k * P3N + i * NPT + n] - sbw[k * 3 + i]);
  }
  __syncthreads();

  // bar_H = 2*(B^T B + I)
  for (int e = t; e < KK * KK; e += nt) {
    int a = e / KK, b2 = e % KK;
    float s = 0.f;
    for (int p = 0; p < P3N; ++p) s += sBarB[p * KK + a] * sBarB[p * KK + b2];
    sH[e] = 2.f * (s + ((a == b2) ? 1.f : 0.f));
  }
  __syncthreads();

  if (t == 0) {
    // Gauss-Jordan inverse of SPD 10x10
    float A[KK][2 * KK];
    for (int r = 0; r < KK; ++r) {
      for (int c = 0; c < KK; ++c) { A[r][c] = sH[r * KK + c]; A[r][KK + c] = (r == c) ? 1.f : 0.f; }
    }
    for (int col = 0; col < KK; ++col) {
      float inv = 1.f / A[col][col];
      for (int c = 0; c < 2 * KK; ++c) A[col][c] *= inv;
      for (int r = 0; r < KK; ++r) {
        if (r == col) continue;
        float f = A[r][col];
        for (int c = 0; c < 2 * KK; ++c) A[r][c] -= f * A[col][c];
      }
    }
    for (int r = 0; r < KK; ++r)
      for (int c = 0; c < KK; ++c) sHinv[r * KK + c] = A[r][KK + c];
    float Ht[KK]; float den = 0.f;
    for (int k = 0; k < KK; ++k) {
      float s = 0.f;
      for (int j = 0; j < KK; ++j) s += sHinv[k * KK + j];
      Ht[k] = s; den += s;
    }
    for (int k = 0; k < KK; ++k) sg[k] = Ht[k] / den;
    for (int r = 0; r < KK; ++r)
      for (int c = 0; c < KK; ++c) sG[r * KK + c] = sHinv[r * KK + c] - Ht[r] * Ht[c] / den;
  }
  __syncthreads();

  // GBt[k][p] = sum_j G[k][j]*barB[p][j]
  for (int e = t; e < KK * P3N; e += nt) {
    int k = e / P3N, p = e % P3N;
    float s = 0.f;
    for (int j = 0; j < KK; ++j) s += sG[k * KK + j] * sBarB[p * KK + j];
    sGBt[e] = s;
    ws[GBT_OFF + e] = s;
  }
  __syncthreads();

  // Mpad[320][312]: rows 0..299 = 2*barB*GBt - I, rows 300..309 = 2*GBt, rest 0
  for (int e = t; e < MRP * KPAD; e += nt) {
    int r = e / KPAD, c = e % KPAD;
    float val = 0.f;
    if (c < P3N) {
      if (r < P3N) {
        float s = 0.f;
        for (int k = 0; k < KK; ++k) s += sBarB[r * KK + k] * sGBt[k * P3N + c];
        val = 2.f * s - ((r == c) ? 1.f : 0.f);
      } else if (r < MR) {
        val = 2.f * sGBt[(r - P3N) * P3N + c];   // 2*sqrt(LAM)*GBt, LAM=1
      }
    }
    ws[M_OFF + e] = val;
  }
  // h[320]
  for (int e = t; e < MRP; e += nt) {
    float val = 0.f;
    if (e < P3N) {
      float s = 0.f;
      for (int k = 0; k < KK; ++k) s += sBarB[e * KK + k] * sg[k];
      val = s;
    } else if (e < MR) val = sg[e - P3N];        // sqrt(LAM)*g
    ws[H_OFF + e] = val;
  }
  for (int e = t; e < KK; e += nt)     ws[G_OFF + e]  = sg[e];
  for (int e = t; e < 3 * KK; e += nt) ws[BW_OFF + e] = sbw[e];
  for (int e = t; e < NPT; e += nt)    ws[SW_OFF + e] = ssw[e];
  if (t == 0) ws[WSUM_OFF] = swsum;
}

// ---------------------------------------------------------------------------
// Kernel 2: per-batch solve. One wave (32 lanes) per batch item.
// ---------------------------------------------------------------------------
__global__ void __launch_bounds__(WAVES * 32)
pace_main(const float* __restrict__ y, const float* __restrict__ ws,
          float* __restrict__ out, int B) {
  __shared__ float bary[WAVES][3][104];
  __shared__ float Wl[WAVES][KPAD * 16];   // W[p][c]: B-matrix of E = M @ W
  __shared__ float At[WAVES][16][16];
  __shared__ float Qs[WAVES][16][16];
  __shared__ float ry[WAVES][3][104];
  __shared__ float Rl[WAVES][9];
  __shared__ float cl[WAVES][KK];
  __shared__ float tl[WAVES][3];
  __shared__ float ywl[WAVES][3];

  const int tid = threadIdx.x;
  const int wid = tid >> 5;
  const int lane = tid & 31;
  int b = blockIdx.x * WAVES + wid;
  if (b >= B) b = B - 1;                       // duplicate work, identical writes

  const float* Mp = ws + M_OFF;
  const float* hp = ws + H_OFF;
  const float* sw = ws + SW_OFF;
  const float wsum = ws[WSUM_OFF];
  const float* yb = y + (size_t)b * P3N;

  // ---- y_w and bar_y -------------------------------------------------------
  float ywr[3];
  for (int j = 0; j < 3; ++j) {
    float s = 0.f;
    for (int n = lane; n < NPT; n += 32) {
      float swv = sw[n];
      s += yb[j * NPT + n] * swv * swv;        // w[n] = sw[n]^2
    }
    for (int off = 16; off > 0; off >>= 1) s += __shfl_down(s, off, 32);
    s = __shfl(s, 0, 32);
    ywr[j] = s / wsum;
  }
  if (lane < 3) ywl[wid][lane] = ywr[lane];
  for (int e = lane; e < 3 * 104; e += 32) {
    int j = e / 104, n = e % 104;
    float v = 0.f;
    if (n < NPT) v = sw[n] * (yb[j * NPT + n] - ywr[j]);
    bary[wid][j][n] = v;
  }
  __syncthreads();

  // ---- materialize W once: W[3n+a', 3a+j] = (a'==a) * bar_y[j][n] ---------
  for (int e = lane; e < KPAD * 16; e += 32) {
    int p = e >> 4, c = e & 15;
    float v = 0.f;
    if (c < 9 && (p % 3) == (c / 3)) v = bary[wid][c % 3][p / 3];
    Wl[wid][e] = v;
  }
  __syncthreads();

  // ---- E = M @ W, Q = [h|E]^T [h|E] via f32 WMMA --------------------------
  const int c16 = lane & 15;
  const int hh  = lane >> 4;                    // half-wave selector
  const float* wbase = &Wl[wid][2 * hh * 16 + c16];  // + kp*64 (+16 for v=1)

  v8f Qacc = {};
  for (int mt = 0; mt < 20; ++mt) {
    const int m0 = mt * 16;
    const float* Arow = Mp + (size_t)(m0 + c16) * KPAD + 2 * hh;
    __builtin_prefetch(Arow + KPAD, 0, 0);      // global_prefetch_b8 next row
    v8f acc = {};
#pragma unroll 6
    for (int kp = 0; kp < 78; ++kp) {
      v2f aF = *(const v2f*)(Arow + kp * 4);    // A[m=c16][k0..k0+1]
      v2f bF = { wbase[kp * 64], wbase[kp * 64 + 16] };
      acc = __builtin_amdgcn_wmma_f32_16x16x4_f32(false, aF, false, bF,
                                                  (short)0, acc, false, false);
    }
    // stage Atile = [h | E | 0]  (16 rows x 16 cols) in LDS
    if (c16 == 15) {
#pragma unroll
      for (int v = 0; v < 8; ++v) At[wid][v + 8 * hh][0] = hp[m0 + v + 8 * hh];
    } else {
      const int colA = c16 + 1;
#pragma unroll
      for (int v = 0; v < 8; ++v)
        At[wid][v + 8 * hh][colA] = (c16 < 9) ? acc[v] : 0.f;
    }
    __syncthreads();
    // Gram: Q += Atile^T * Atile  (A-frag == B-frag)
#pragma unroll
    for (int kk2 = 0; kk2 < 16; kk2 += 4) {
      float f0 = At[wid][kk2 + 2 * hh + 0][c16];
      float f1 = At[wid][kk2 + 2 * hh + 1][c16];
      v2f fF = {f0, f1};
      Qacc = __builtin_amdgcn_wmma_f32_16x16x4_f32(false, fF, false, fF,
                                                   (short)0, Qacc, false, false);
    }
    __syncthreads();
  }
#pragma unroll
  for (int v = 0; v < 8; ++v) Qs[wid][v + 8 * hh][c16] = Qacc[v];
  __syncthreads();

  // ---- smallest eigenvector of 10x10 PSD Q: ridge Cholesky + inv. iter ----
  if (lane == 0) {
    float Am[KK][KK], L[KK][KK], vv[KK], yv[KK], xv[KK];
    float tr = 0.f;
    for (int r = 0; r < KK; ++r) {
      for (int c = 0; c < KK; ++c) Am[r][c] = Qs[wid][r][c];
      tr += Am[r][r];
    }
    float eps = 1e-6f * tr + 1e-18f;
    for (int i = 0; i < KK; ++i) Am[i][i] += eps;
    for (int j = 0; j < KK; ++j) {
      float s = Am[j][j];
      for (int k = 0; k < j; ++k) s -= L[j][k] * L[j][k];
      float d = sqrtf(fmaxf(s, 1e-30f));
      L[j][j] = d;
      for (int i = j + 1; i < KK; ++i) {
        float s2 = Am[i][j];
        for (int k = 0; k < j; ++k) s2 -= L[i][k] * L[j][k];
        L[i][j] = s2 / d;
      }
    }
    for (int i = 0; i < KK; ++i) vv[i] = 1.f + 0.01f * (float)i;
    for (int it = 0; it < 20; ++it) {
      for (int i = 0; i < KK; ++i) {
        float s = vv[i];
        for (int k = 0; k < i; ++k) s -= L[i][k] * yv[k];
        yv[i] = s / L[i][i];
      }
      for (int i = KK - 1; i >= 0; --i) {
        float s = yv[i];
        for (int k = i + 1; k < KK; ++k) s -= L[k][i] * xv[k];
        xv[i] = s / L[i][i];
      }
      float nrm = 0.f;
      for (int i = 0; i < KK; ++i) nrm += xv[i] * xv[i];
      float inv = rsqrtf(fmaxf(nrm, 1e-30f));
      for (int i = 0; i < KK; ++i) vv[i] = xv[i] * inv;
    }
    float inv0 = 1.f / vv[0];
    // R[r][a] = v[1+3a+r]/v[0], stored row-major
    for (int r = 0; r < 3; ++r)
      for (int a = 0; a < 3; ++a) Rl[wid][3 * r + a] = vv[1 + 3 * a + r] * inv0;
  }
  __syncthreads();

  // ---- ry[a][n] = sum_r R[r][a]*bar_y[r][n] -------------------------------
  for (int e = lane; e < 3 * 104; e += 32) {
    int a = e / 104, n = e % 104;
    float s = 0.f;
    if (n < NPT)
      for (int r = 0; r < 3; ++r) s += Rl[wid][3 * r + a] * bary[wid][r][n];
    ry[wid][a][n] = s;
  }
  __syncthreads();

  // ---- c[k] = 2*sum_q GBt[k][q]*ry[q%3][q/3] + g[k] -----------------------
  if (lane < KK) {
    float s = ws[G_OFF + lane];
    const float* gb = ws + GBT_OFF + lane * P3N;
    for (int n = 0; n < NPT; ++n)
      for (int a = 0; a < 3; ++a)
        s += 2.f * gb[3 * n + a] * ry[wid][a][n];
    cl[wid][lane] = s;
  }
  __syncthreads();

  // ---- t = y_w - R @ (b_w^T c) --------------------------------------------
  if (lane == 0) {
    float sv[3] = {0.f, 0.f, 0.f};
    for (int k = 0; k < KK; ++k)
      for (int a2 = 0; a2 < 3; ++a2) sv[a2] += ws[BW_OFF + k * 3 + a2] * cl[wid][k];
    for (int a = 0; a < 3; ++a) {
      float tv = ywl[wid][a];
      for (int a2 = 0; a2 < 3; ++a2) tv -= Rl[wid][3 * a + a2] * sv[a2];
      tl[wid][a] = tv;
    }
  }
  __syncthreads();

  // ---- outputs: R (B*9) | t (B*3) | c (B*10), concatenated flat -----------
  if (lane < 9)  out[(size_t)b * 9 + lane] = Rl[wid][lane];
  if (lane < 3)  out[(size_t)B * 9 + (size_t)b * 3 + lane] = tl[wid][lane];
  if (lane < KK) out[(size_t)B * 12 + (size_t)b * KK + lane] = cl[wid][lane];
}

// ---------------------------------------------------------------------------
extern "C" void kernel_launch(void* const* d_in, const int* in_sizes, int n_in,
                              void* d_out, int out_size, void* d_ws, size_t ws_size,
                              hipStream_t stream) {
  const float* y  = (const float*)d_in[0];
  const float* w  = (const float*)d_in[1];
  const float* mk = (const float*)d_in[2];
  float* ws = (float*)d_ws;
  float* out = (float*)d_out;
  const int B = in_sizes[0] / P3N;

  pace_precompute<<<1, 256, 0, stream>>>(w, mk, ws);
  pace_main<<<(B + WAVES - 1) / WAVES, WAVES * 32, 0, stream>>>(y, ws, out, B);
}